// Net_63797444214872
// MI455X (gfx1250) — compile-verified
//
#include <hip/hip_runtime.h>
#include <hip/hip_bf16.h>

typedef float v2f __attribute__((ext_vector_type(2)));
typedef float v8f __attribute__((ext_vector_type(8)));

#define THREADS 256

// ---------------------------------------------------------------------------
// Pass 0: deg[v] = 1 (self-loop weight), zero the 2-wide logit accumulator.
// ---------------------------------------------------------------------------
__global__ void k_init(float* __restrict__ deg, float* __restrict__ s2, int n) {
    int i = blockIdx.x * blockDim.x + threadIdx.x;
    if (i < n) deg[i] = 1.0f;
    if (i < 2) s2[i] = 0.0f;
}

// ---------------------------------------------------------------------------
// Pass A: deg[dst] += w  (float atomics; node array is L2-resident, 2MB).
// 4 edges per thread via B128 loads.
// ---------------------------------------------------------------------------
__global__ void k_edge_deg(const int* __restrict__ dst, const float* __restrict__ ew,
                           float* __restrict__ deg, long long E) {
    long long i = (long long)blockIdx.x * blockDim.x + threadIdx.x;
    long long base = i * 4;
    if (base + 3 < E) {
        int4   d = ((const int4*)dst)[i];
        float4 w = ((const float4*)ew)[i];
        atomicAdd(&deg[d.x], w.x);
        atomicAdd(&deg[d.y], w.y);
        atomicAdd(&deg[d.z], w.z);
        atomicAdd(&deg[d.w], w.w);
    } else {
        for (long long e = base; e < E && e < base + 4; ++e)
            atomicAdd(&deg[dst[e]], ew[e]);
    }
}

// ---------------------------------------------------------------------------
// Pass B: dis[v] = rsqrt(deg) (in place); seed a,g with the self-loop term
//   self-loop norm = dis[v]*1*dis[v] = 1/deg[v]
//   a[v] = x[v]/deg[v], g[v] = 1/deg[v]
// ---------------------------------------------------------------------------
__global__ void k_node_norm(const float* __restrict__ x, float* __restrict__ degdis,
                            float* __restrict__ a, float* __restrict__ g, int n) {
    int i = blockIdx.x * blockDim.x + threadIdx.x;
    if (i < n) {
        float d   = degdis[i];          // >= 1 always
        float inv = 1.0f / d;
        degdis[i] = rsqrtf(d);          // becomes dis[]
        a[i]      = x[i] * inv;
        g[i]      = inv;
    }
}

// ---------------------------------------------------------------------------
// Pass C: per edge, norm = dis[s]*w*dis[d];
//   a[d] += norm * x[s]   (layer-1 message, scalar since x is [N,1])
//   g[s] += norm          (layer-2 collapses to out-norm sums per node)
// Gathers and atomics hit L2-resident 2MB arrays.
// ---------------------------------------------------------------------------
__device__ __forceinline__ void edge_msg_one(int s, int d, float w,
                                             const float* __restrict__ x,
                                             const float* __restrict__ dis,
                                             float* __restrict__ a,
                                             float* __restrict__ g) {
    float nm = dis[s] * w * dis[d];
    atomicAdd(&a[d], nm * x[s]);
    atomicAdd(&g[s], nm);
}

__global__ void k_edge_msg(const int* __restrict__ src, const int* __restrict__ dst,
                           const float* __restrict__ ew, const float* __restrict__ x,
                           const float* __restrict__ dis, float* __restrict__ a,
                           float* __restrict__ g, long long E) {
    long long i = (long long)blockIdx.x * blockDim.x + threadIdx.x;
    long long base = i * 4;
    if (base + 3 < E) {
        int4   s = ((const int4*)src)[i];
        int4   d = ((const int4*)dst)[i];
        float4 w = ((const float4*)ew)[i];
        edge_msg_one(s.x, d.x, w.x, x, dis, a, g);
        edge_msg_one(s.y, d.y, w.y, x, dis, a, g);
        edge_msg_one(s.z, d.z, w.z, x, dis, a, g);
        edge_msg_one(s.w, d.w, w.w, x, dis, a, g);
    } else {
        for (long long e = base; e < E && e < base + 4; ++e)
            edge_msg_one(src[e], dst[e], ew[e], x, dis, a, g);
    }
}

// ---------------------------------------------------------------------------
// Pass D: acc16[c] = sum_v g[v] * relu(a[v]*W1[c] + b1[c]);  s2 = acc16 @ W2.
// Per-thread 16-wide accumulators -> wave32 butterfly reduce -> per-wave
// f32 WMMA (16x16x4, K chained over 4 chunks) computes the 16x2 GEMV exactly,
// then lanes 0,1 atomically add the two logit partials.
// f32 WMMA is mandatory here: acc16 ~ O(1e5..1e6) overflows f16 and loses all
// logit precision in bf16.
// ---------------------------------------------------------------------------
__global__ void k_nodes_acc(const float* __restrict__ a, const float* __restrict__ g,
                            const float* __restrict__ W1, const float* __restrict__ b1,
                            const float* __restrict__ W2, float* __restrict__ s2, int n) {
    float w1r[16], b1r[16], acc[16];
#pragma unroll
    for (int c = 0; c < 16; ++c) { w1r[c] = W1[c]; b1r[c] = b1[c]; acc[c] = 0.0f; }

    int stride = gridDim.x * blockDim.x;
    for (int v = blockIdx.x * blockDim.x + threadIdx.x; v < n; v += stride) {
        float av = a[v];
        float gv = g[v];
#pragma unroll
        for (int c = 0; c < 16; ++c)
            acc[c] += gv * fmaxf(fmaf(av, w1r[c], b1r[c]), 0.0f);
    }

    // wave32 butterfly reduction; all lanes end holding the full wave sums
#pragma unroll
    for (int c = 0; c < 16; ++c) {
        acc[c] += __shfl_xor(acc[c], 16, 32);
        acc[c] += __shfl_xor(acc[c],  8, 32);
        acc[c] += __shfl_xor(acc[c],  4, 32);
        acc[c] += __shfl_xor(acc[c],  2, 32);
        acc[c] += __shfl_xor(acc[c],  1, 32);
    }

    int  lane = threadIdx.x & 31;
    bool hi   = lane >= 16;
    int  ncol = lane & 15;

#if defined(__has_builtin) && __has_builtin(__builtin_amdgcn_wmma_f32_16x16x4_f32)
    // A(16x4): row-broadcast of acc chunk. Lane layout (ISA 7.12.2, 32-bit A 16x4):
    //   lanes 0-15 hold K=0,1 in v[0],v[1]; lanes 16-31 hold K=2,3.
    // B(4x16): B[k,n] = W2[chunk*4+k][n] for n<2 else 0; K split mirrors A.
    // D[0,n] lands in c8[0] of lane n -> lanes 0,1 hold the two logit partials.
    v8f c8 = {};
#pragma unroll
    for (int ch = 0; ch < 4; ++ch) {
        float aX = hi ? acc[ch * 4 + 2] : acc[ch * 4 + 0];
        float aY = hi ? acc[ch * 4 + 3] : acc[ch * 4 + 1];
        int   k0 = ch * 4 + (hi ? 2 : 0);
        float bX = (ncol < 2) ? W2[k0 * 2 + ncol]       : 0.0f;
        float bY = (ncol < 2) ? W2[(k0 + 1) * 2 + ncol] : 0.0f;
        v2f A = {aX, aY};
        v2f B = {bX, bY};
        c8 = __builtin_amdgcn_wmma_f32_16x16x4_f32(false, A, false, B,
                                                   (short)0, c8, false, false);
    }
    if (lane < 2) atomicAdd(&s2[lane], c8[0]);
#else
    if (lane < 2) {
        float sv = 0.0f;
#pragma unroll
        for (int k = 0; k < 16; ++k) sv += acc[k] * W2[k * 2 + lane];
        atomicAdd(&s2[lane], sv);
    }
#endif
}

// ---------------------------------------------------------------------------
// Finalize: s += N*b2, softmax over 2 logits.
// ---------------------------------------------------------------------------
__global__ void k_finalize(const float* __restrict__ s2, const float* __restrict__ b2,
                           float fN, float* __restrict__ out) {
    float s0 = s2[0] + fN * b2[0];
    float s1 = s2[1] + fN * b2[1];
    float m  = fmaxf(s0, s1);
    float e0 = expf(s0 - m);
    float e1 = expf(s1 - m);
    float inv = 1.0f / (e0 + e1);
    out[0] = e0 * inv;
    out[1] = e1 * inv;
}

extern "C" void kernel_launch(void* const* d_in, const int* in_sizes, int n_in,
                              void* d_out, int out_size, void* d_ws, size_t ws_size,
                              hipStream_t stream) {
    const float* x  = (const float*)d_in[0];   // [N,1]
    const int*   ei = (const int*)d_in[1];     // [2,E] row-major (int32)
    const float* ew = (const float*)d_in[2];   // [E]
    const float* W1 = (const float*)d_in[3];   // [1,16]
    const float* b1 = (const float*)d_in[4];   // [16]
    const float* W2 = (const float*)d_in[5];   // [16,2]
    const float* b2 = (const float*)d_in[6];   // [2]

    int       n = in_sizes[0];                 // 500,000 nodes
    long long E = in_sizes[2];                 // 16,000,000 edges

    const int* src = ei;
    const int* dst = ei + E;

    float* ws  = (float*)d_ws;
    float* deg = ws;                           // [n] deg -> dis (in place)
    float* aa  = ws + (size_t)n;               // [n] layer-1 scalar aggregate
    float* gg  = ws + 2 * (size_t)n;           // [n] out-norm sums
    float* s2  = ws + 3 * (size_t)n;           // [2] logit accumulator

    int nb_nodes = (n + THREADS - 1) / THREADS;
    long long e4 = (E + 3) / 4;
    int nb_edges = (int)((e4 + THREADS - 1) / THREADS);

    k_init<<<nb_nodes, THREADS, 0, stream>>>(deg, s2, n);
    k_edge_deg<<<nb_edges, THREADS, 0, stream>>>(dst, ew, deg, E);
    k_node_norm<<<nb_nodes, THREADS, 0, stream>>>(x, deg, aa, gg, n);
    k_edge_msg<<<nb_edges, THREADS, 0, stream>>>(src, dst, ew, x, deg, aa, gg, E);
    k_nodes_acc<<<1024, THREADS, 0, stream>>>(aa, gg, W1, b1, W2, s2, n);
    k_finalize<<<1, 1, 0, stream>>>(s2, b2, (float)n, (float*)d_out);
}